// VirtualCell_6347961664101
// MI455X (gfx1250) — compile-verified
//
#include <hip/hip_runtime.h>

typedef float v2f __attribute__((ext_vector_type(2)));
typedef float v8f __attribute__((ext_vector_type(8)));

#define BATCH 512
#define CELL  768
#define SEQ   1000
#define RANK  64
#define K1    (SEQ * 4)          // 4000
#define DD    256
#define NW    (2 * CELL * RANK)  // 98304
#define HALF  (CELL * RANK)      // 49152
#define JCHUNKS 16
#define JPER    (CELL / JCHUNKS) // 48
#define BSUB  4                  // 4 x 16 = 64 batches per workgroup
#define NPART (JCHUNKS * 4)      // 64 per-wave t-partial slots

// ---------------------------------------------------------------------------
// Kernel A: F = relu(dna @ W_dna + b_dna)   (512 x 4000)*(4000 x 256)
// One wave computes one 16x16 f32 tile with V_WMMA_F32_16X16X4_F32, K-loop 1000.
// ---------------------------------------------------------------------------
__global__ __launch_bounds__(128) void k_dna(
    const float* __restrict__ dna, const float* __restrict__ Wd,
    const float* __restrict__ bd, float* __restrict__ F)
{
    const int tid   = threadIdx.x;
    const int lane  = tid & 31;
    const int wave  = tid >> 5;
    const int btile = blockIdx.x;
    const int ntile = blockIdx.y * 4 + wave;
    const int n0    = ntile * 16;
    const int mlane = lane & 15;
    const int kl    = (lane >> 4) * 2;    // A/B K-pair select per ISA 16x4 layout

    const float* arow = dna + (size_t)(btile * 16 + mlane) * K1;
    v8f c = {};
    for (int k0 = 0; k0 < K1; k0 += 4) {
        v2f a = *(const v2f*)(arow + k0 + kl);
        const float* bp = Wd + (size_t)(k0 + kl) * DD + n0 + mlane;
        v2f bfrag;
        bfrag.x = bp[0];
        bfrag.y = bp[DD];
        c = __builtin_amdgcn_wmma_f32_16x16x4_f32(false, a, false, bfrag,
                                                  (short)0, c, false, false);
    }
    const int   n     = n0 + mlane;
    const float bias  = bd[n];
    const int   mbase = btile * 16 + ((lane >> 4) << 3);
#pragma unroll
    for (int v = 0; v < 8; ++v) {
        float val = c[v] + bias;
        F[(size_t)(mbase + v) * DD + n] = val > 0.f ? val : 0.f;
    }
}

// ---------------------------------------------------------------------------
// Kernel B (phase 1, fused): V = F @ Wv + bv contracted on the fly with x[b,j]
// into per-wave register t-partials.  64 batches/WG: each Wv fragment load
// feeds 4 independent WMMA accumulator chains (4x B-operand reuse, better
// latency hiding).  rt is the outer loop so each lane's (b,r) set is fixed
// across j -> register accumulation, single store per partial (deterministic).
// Grid: (512/64 = 8 batch groups, 16 j-chunks of 48).  4 waves/WG.
// ---------------------------------------------------------------------------
__global__ __launch_bounds__(128) void k_vphase(
    const float* __restrict__ F, const float* __restrict__ Wh,
    const float* __restrict__ bh, const float* __restrict__ x,
    float* __restrict__ tpart)
{
    __shared__ float Fs[BSUB * 16 * DD];     // 64 KB : F tile, 64 batches
    __shared__ float xs[BSUB * 16 * JPER];   // 12 KB : x tile, 64 b x 48 j

    const int tid  = threadIdx.x;
    const int lane = tid & 31;
    const int wave = tid >> 5;
    const int bg   = blockIdx.x;             // batch group of 64
    const int jc   = blockIdx.y;

    {   // stage F (64x256) and x (64x48) into LDS
        const float4* Fsrc = (const float4*)(F + (size_t)bg * 64 * DD);
        float4*       Fdst = (float4*)Fs;
        for (int i = tid; i < 64 * DD / 4; i += 128) Fdst[i] = Fsrc[i];
        for (int i = tid; i < 64 * JPER; i += 128) {
            int b = i / JPER, jl = i % JPER;
            xs[i] = x[(size_t)(bg * 64 + b) * CELL + jc * JPER + jl];
        }
    }
    __syncthreads();

    const int mlane = lane & 15;
    const int kl    = (lane >> 4) * 2;
    const int hi    = lane >> 4;

    for (int rt = 0; rt < 4; ++rt) {
        // this wave+rt owns (b, r) with r = rt*16 + mlane, b = s*16 + hi*8 + v
        float acc[BSUB][8];
#pragma unroll
        for (int s = 0; s < BSUB; ++s)
#pragma unroll
            for (int v = 0; v < 8; ++v) acc[s][v] = 0.f;

        for (int jl = wave; jl < JPER; jl += 4) {
            const int j    = jc * JPER + jl;
            const int ncol = HALF + j * RANK + rt * 16 + mlane;
            v8f c[BSUB] = {};
            for (int k0 = 0; k0 < DD; k0 += 4) {
                const float* bp = Wh + (size_t)(k0 + kl) * NW + ncol;
                v2f bfrag;
                bfrag.x = bp[0];
                bfrag.y = bp[NW];
#pragma unroll
                for (int s = 0; s < BSUB; ++s) {
                    v2f a = *(const v2f*)(Fs + (s * 16 + mlane) * DD + k0 + kl);
                    c[s] = __builtin_amdgcn_wmma_f32_16x16x4_f32(
                        false, a, false, bfrag, (short)0, c[s], false, false);
                }
            }
            const float bias = bh[ncol];
#pragma unroll
            for (int s = 0; s < BSUB; ++s)
#pragma unroll
                for (int v = 0; v < 8; ++v) {
                    int b = s * 16 + hi * 8 + v;
                    acc[s][v] += (c[s][v] + bias) * xs[b * JPER + jl];
                }
        }
        // single deterministic store of this wave's partial slot
        const int slot = jc * 4 + wave;
        const int r    = rt * 16 + mlane;
#pragma unroll
        for (int s = 0; s < BSUB; ++s)
#pragma unroll
            for (int v = 0; v < 8; ++v) {
                int b = bg * 64 + s * 16 + hi * 8 + v;
                tpart[((size_t)slot * BATCH + b) * RANK + r] = acc[s][v];
            }
    }
}

// ---------------------------------------------------------------------------
// Kernel C: reduce the 64 per-wave partial slots -> t[b,r]  (fixed order)
// ---------------------------------------------------------------------------
__global__ void k_treduce(const float* __restrict__ tpart, float* __restrict__ t)
{
    int idx = blockIdx.x * blockDim.x + threadIdx.x;
    if (idx >= BATCH * RANK) return;
    float s = 0.f;
    for (int p = 0; p < NPART; ++p)
        s += tpart[(size_t)p * BATCH * RANK + idx];
    t[idx] = s;
}

// ---------------------------------------------------------------------------
// Kernel D (phase 2, fused): U = F @ Wu + bu contracted with t[b,r].
// 64 batches/WG (4 sub-tiles sharing each Wu fragment), half-wave shfl_xor
// reduction over r -> pred[b,i].
// Grid: (8 batch groups, 48 i-chunks of 16).  4 waves/WG, 4 i per wave.
// ---------------------------------------------------------------------------
__global__ __launch_bounds__(128) void k_uphase(
    const float* __restrict__ F, const float* __restrict__ Wh,
    const float* __restrict__ bh, const float* __restrict__ t,
    float* __restrict__ pred)
{
    __shared__ float Fs[BSUB * 16 * DD];     // 64 KB
    __shared__ float ts[BSUB * 16 * RANK];   // 16 KB

    const int tid  = threadIdx.x;
    const int lane = tid & 31;
    const int wave = tid >> 5;
    const int bg   = blockIdx.x;
    const int ic   = blockIdx.y;

    {
        const float4* Fsrc = (const float4*)(F + (size_t)bg * 64 * DD);
        float4*       Fdst = (float4*)Fs;
        for (int i = tid; i < 64 * DD / 4; i += 128) Fdst[i] = Fsrc[i];
        const float4* Tsrc = (const float4*)(t + (size_t)bg * 64 * RANK);
        float4*       Tdst = (float4*)ts;
        for (int i = tid; i < 64 * RANK / 4; i += 128) Tdst[i] = Tsrc[i];
    }
    __syncthreads();

    const int mlane = lane & 15;
    const int kl    = (lane >> 4) * 2;
    const int hi    = lane >> 4;

    for (int il = wave; il < 16; il += 4) {
        const int i = ic * 16 + il;
        float acc[BSUB][8];
#pragma unroll
        for (int s = 0; s < BSUB; ++s)
#pragma unroll
            for (int v = 0; v < 8; ++v) acc[s][v] = 0.f;

        for (int rt = 0; rt < 4; ++rt) {
            const int ncol = i * RANK + rt * 16 + mlane;   // U half: n = i*64 + r
            v8f c[BSUB] = {};
            for (int k0 = 0; k0 < DD; k0 += 4) {
                const float* bp = Wh + (size_t)(k0 + kl) * NW + ncol;
                v2f bfrag;
                bfrag.x = bp[0];
                bfrag.y = bp[NW];
#pragma unroll
                for (int s = 0; s < BSUB; ++s) {
                    v2f a = *(const v2f*)(Fs + (s * 16 + mlane) * DD + k0 + kl);
                    c[s] = __builtin_amdgcn_wmma_f32_16x16x4_f32(
                        false, a, false, bfrag, (short)0, c[s], false, false);
                }
            }
            const float bias = bh[ncol];
            const int   r    = rt * 16 + mlane;
#pragma unroll
            for (int s = 0; s < BSUB; ++s)
#pragma unroll
                for (int v = 0; v < 8; ++v)
                    acc[s][v] += (c[s][v] + bias) * ts[(s * 16 + hi * 8 + v) * RANK + r];
        }
        // reduce over the 16 r-lanes within each half-wave (hi bit untouched)
#pragma unroll
        for (int s = 0; s < BSUB; ++s)
#pragma unroll
            for (int v = 0; v < 8; ++v) {
                float a = acc[s][v];
                a += __shfl_xor(a, 1, 32);
                a += __shfl_xor(a, 2, 32);
                a += __shfl_xor(a, 4, 32);
                a += __shfl_xor(a, 8, 32);
                if (mlane == 0)
                    pred[(size_t)(bg * 64 + s * 16 + hi * 8 + v) * CELL + i] = a;
            }
    }
}

// ---------------------------------------------------------------------------
extern "C" void kernel_launch(void* const* d_in, const int* in_sizes, int n_in,
                              void* d_out, int out_size, void* d_ws, size_t ws_size,
                              hipStream_t stream)
{
    (void)in_sizes; (void)n_in; (void)out_size; (void)ws_size;

    const float* x   = (const float*)d_in[0];  // control_state        (512, 768)
    const float* dna = (const float*)d_in[1];  // sequence_prob_matrix (512, 4000)
    const float* Wd  = (const float*)d_in[2];  // W_dna   (4000, 256)
    const float* bd  = (const float*)d_in[3];  // b_dna   (256,)
    const float* Wh  = (const float*)d_in[4];  // W_hyper (256, 98304)
    const float* bh  = (const float*)d_in[5];  // b_hyper (98304,)
    float* pred = (float*)d_out;               // (512, 768)

    // workspace layout: F | tpart | t   (~8.6 MB total)
    float* F     = (float*)d_ws;                               // 512*256
    float* tpart = F + (size_t)BATCH * DD;                     // 64*512*64
    float* t     = tpart + (size_t)NPART * BATCH * RANK;       // 512*64

    k_dna    <<<dim3(BATCH / 16, 4),         128, 0, stream>>>(dna, Wd, bd, F);
    k_vphase <<<dim3(BATCH / 64, JCHUNKS),   128, 0, stream>>>(F, Wh, bh, x, tpart);
    k_treduce<<<(BATCH * RANK + 255) / 256,  256, 0, stream>>>(tpart, t);
    k_uphase <<<dim3(BATCH / 64, CELL / 16), 128, 0, stream>>>(F, Wh, bh, t, pred);
}